// GATTrafficPredictionModel_65936337928659
// MI455X (gfx1250) — compile-verified
//
#include <hip/hip_runtime.h>
#include <math.h>

// ---------------------------------------------------------------------------
// GAT traffic model, collapsed form:
//   h rows are identical  =>  softmax(attention) is uniform  =>  att @ Wh = Wh
//   out[b,c] = bf[c] + sum_{c'} Wv[b,c'] * R[c,c']
//   with  Hc = elu(x_last @ W1),  Wv = Hc @ W_out,  R[c,c'] = sum_n Wf[c, n*64+c']
// Dominant cost: streaming 8.4 MB of Wf once (~0.36 us at 23.3 TB/s).
// GEMMs use V_WMMA_F32_16X16X4_F32 (fp32 in/out, wave32, 16x16 tiles).
// ELU uses hardware v_exp_f32 (branchless select) instead of libm expm1f.
// ---------------------------------------------------------------------------

typedef float v2f __attribute__((ext_vector_type(2)));
typedef float v8f __attribute__((ext_vector_type(8)));

#define B_   32
#define S_   12
#define F_   128
#define H_   64
#define K_   8
#define C_   64
#define N_   512
#define KH_  512   // K_*H_

// workspace layout (float offsets)
#define R_OFF   0
#define HC_OFF  (R_OFF + C_ * C_)          // 4096
#define WV_OFF  (HC_OFF + B_ * KH_)        // 4096 + 16384

// ---------------------------------------------------------------------------
// Kernel 1: R[c,c'] = sum_n Wf[c, n*64 + c'].  64 blocks (one per c) x 256.
// Fully coalesced 1KB reads per iteration; deterministic two-phase reduction.
// ---------------------------------------------------------------------------
__global__ __launch_bounds__(256) void reduce_wf_kernel(const float* __restrict__ Wf,
                                                        float* __restrict__ R) {
  const int c   = blockIdx.x;        // 0..63
  const int tid = threadIdx.x;       // 0..255
  const int c2  = tid & 63;
  const int p   = tid >> 6;          // 0..3
  const float* __restrict__ row = Wf + (size_t)c * (N_ * C_);

  float s = 0.f;
  #pragma unroll 4
  for (int n = p; n < N_; n += 4) {
    __builtin_prefetch(row + (n + 16) * C_ + c2, 0, 1);  // global_prefetch_b8
    s += row[n * C_ + c2];
  }

  __shared__ float part[256];
  part[tid] = s;
  __syncthreads();
  if (tid < 64) {
    R[c * C_ + tid] = (part[tid] + part[64 + tid]) + (part[128 + tid] + part[192 + tid]);
  }
}

// Branchless ELU(alpha=1) using the hardware transcendental (v_exp_f32).
__device__ __forceinline__ float elu_fast(float v) {
  const float e = __expf(v) - 1.0f;   // TRANS32 op, co-executes with VALU
  return v > 0.0f ? v : e;            // v_cndmask, no EXEC branching
}

// ---------------------------------------------------------------------------
// WMMA fragments for V_WMMA_F32_16X16X4_F32 (wave32):
//   A 16x4 : lane<16 -> K={0,1} in v[0],v[1]; lane>=16 -> K={2,3}
//   B 4x16 : mirrored (N = lane&15, K = 2*half + v)
//   C/D    : 8 VGPRs, M = r + 8*half, N = lane&15
// ---------------------------------------------------------------------------

// Kernel 2: U = x_last(32x128) @ W1(128x512), W1[f, k*64+h] = W_heads[k,f,h];
// ELU applied, result -> Hc.  One wave per 16x16 tile, 64 tiles, K=128.
__global__ __launch_bounds__(32) void gemm1_elu_kernel(const float* __restrict__ x,
                                                       const float* __restrict__ W_heads,
                                                       float* __restrict__ Hc) {
  const int mt   = blockIdx.x & 1;    // 0..1  (M tiles: 32/16)
  const int nt   = blockIdx.x >> 1;   // 0..31 (N tiles: 512/16)
  const int lane = threadIdx.x;       // 0..31 (wave32)
  const int r16  = lane & 15;
  const int half = lane >> 4;

  const int m = mt * 16 + r16;        // A row for this lane
  const int j = nt * 16 + r16;        // output column = k*64 + h (tile never crosses head)
  const int khead = j >> 6;
  const int hcol  = j & 63;

  const float* __restrict__ arow = x + m * (S_ * F_) + (S_ - 1) * F_; // x[m, 11, :]
  const float* __restrict__ bcol = W_heads + (size_t)khead * (F_ * H_) + hcol;

  v8f acc = {};
  #pragma unroll 4
  for (int kk = 0; kk < F_; kk += 4) {
    const int f = kk + 2 * half;
    const v2f a = *(const v2f*)(arow + f);        // 8B-aligned (f even)
    v2f b;
    b.x = bcol[(size_t)f * H_];
    b.y = bcol[(size_t)(f + 1) * H_];
    acc = __builtin_amdgcn_wmma_f32_16x16x4_f32(false, a, false, b,
                                                (short)0, acc, false, false);
  }

  #pragma unroll
  for (int r = 0; r < 8; ++r) {
    const int mr = mt * 16 + r + 8 * half;
    Hc[mr * KH_ + j] = elu_fast(acc[r]);
  }
}

// Kernel 3: Wv = Hc(32x512) @ W_out(512x64).  8 tiles (2x4), K=512.
__global__ __launch_bounds__(32) void gemm2_kernel(const float* __restrict__ Hc,
                                                   const float* __restrict__ W_out,
                                                   float* __restrict__ Wv) {
  const int mt   = blockIdx.x & 1;    // 0..1
  const int nt   = blockIdx.x >> 1;   // 0..3
  const int lane = threadIdx.x;
  const int r16  = lane & 15;
  const int half = lane >> 4;

  const int m = mt * 16 + r16;
  const int j = nt * 16 + r16;
  const float* __restrict__ arow = Hc + m * KH_;

  v8f acc = {};
  #pragma unroll 4
  for (int kk = 0; kk < KH_; kk += 4) {
    const int f = kk + 2 * half;
    const v2f a = *(const v2f*)(arow + f);
    v2f b;
    b.x = W_out[f * C_ + j];
    b.y = W_out[(f + 1) * C_ + j];
    acc = __builtin_amdgcn_wmma_f32_16x16x4_f32(false, a, false, b,
                                                (short)0, acc, false, false);
  }

  #pragma unroll
  for (int r = 0; r < 8; ++r) {
    Wv[(mt * 16 + r + 8 * half) * C_ + j] = acc[r];
  }
}

// Kernel 4: out[b,c] = bf[c] + sum_{c'} Wv[b,c'] * R[c,c'].  2048 outputs.
__global__ __launch_bounds__(256) void final_kernel(const float* __restrict__ Wv,
                                                    const float* __restrict__ R,
                                                    const float* __restrict__ bf,
                                                    float* __restrict__ out) {
  const int gid = blockIdx.x * 256 + threadIdx.x;  // 0..2047
  const int b = gid >> 6;
  const int c = gid & 63;
  const float* __restrict__ wv = Wv + b * C_;
  const float* __restrict__ rr = R + c * C_;
  float s = bf[c];
  #pragma unroll 8
  for (int i = 0; i < C_; ++i) s = fmaf(wv[i], rr[i], s);
  out[gid] = s;
}

// ---------------------------------------------------------------------------
extern "C" void kernel_launch(void* const* d_in, const int* in_sizes, int n_in,
                              void* d_out, int out_size, void* d_ws, size_t ws_size,
                              hipStream_t stream) {
  (void)in_sizes; (void)n_in; (void)out_size; (void)ws_size;

  const float* x       = (const float*)d_in[0];
  const float* W_heads = (const float*)d_in[1];
  // d_in[2] (a1_heads), d_in[3] (a2_heads): provably no effect on output
  const float* W_out   = (const float*)d_in[4];
  // d_in[5] (a1_out), d_in[6] (a2_out): provably no effect on output
  const float* Wf      = (const float*)d_in[7];
  const float* bf      = (const float*)d_in[8];

  float* out = (float*)d_out;
  float* ws  = (float*)d_ws;
  float* R   = ws + R_OFF;    //  64x64
  float* Hc  = ws + HC_OFF;   //  32x512
  float* Wv  = ws + WV_OFF;   //  32x64

  reduce_wf_kernel<<<64, 256, 0, stream>>>(Wf, R);       // the 8.4 MB stream
  gemm1_elu_kernel<<<64, 32, 0, stream>>>(x, W_heads, Hc);
  gemm2_kernel<<<8, 32, 0, stream>>>(Hc, W_out, Wv);
  final_kernel<<<8, 256, 0, stream>>>(Wv, R, bf, out);
}